// LieGroupEquivariantLayer_2405181686613
// MI455X (gfx1250) — compile-verified
//
#include <hip/hip_runtime.h>

typedef __attribute__((ext_vector_type(2))) float v2f;
typedef __attribute__((ext_vector_type(8))) float v8f;

#define Q_ 4
#define D_ 7
#define G_ 4
#define C_ 16
#define X_ 2
#define N_ 64
#define KDIM (N_ * Q_ * D_)   // 1792 = contraction (j,l,s)
#define NCOL (Q_ * C_ * D_)   // 448  = output cols (q,d,r)

struct c32 { float re, im; };
__device__ __forceinline__ c32 cmul(c32 a, c32 b) {
    return { a.re * b.re - a.im * b.im, a.re * b.im + a.im * b.re };
}
__device__ __forceinline__ c32 cadd(c32 a, c32 b) { return { a.re + b.re, a.im + b.im }; }
__device__ __forceinline__ c32 cload(const float* p) { return { p[0], p[1] }; }
__device__ __forceinline__ void cstore(float* p, c32 a) { p[0] = a.re; p[1] = a.im; }

// ---------------------------------------------------------------------------
// k0: cgf[q,r,s,t] = sum_g f[q,g] * cg[g, l=1, s, m=1, t, q, r]   (252 complex)
// ---------------------------------------------------------------------------
__global__ void k_cgf(const float* __restrict__ f, const float* __restrict__ cg,
                      float* __restrict__ cgf) {
    int idx = blockIdx.x * blockDim.x + threadIdx.x;
    if (idx >= Q_ * D_ * 3 * 3) return;
    int t = idx % 3, s = (idx / 3) % 3, r = (idx / 9) % D_, q = idx / 63;
    c32 acc = {0.f, 0.f};
    for (int g = 0; g < G_; ++g) {
        long cgi = ((((((long)g * Q_ + 1) * D_ + s) * Q_ + 1) * D_ + t) * Q_ + q) * D_ + r;
        acc = cadd(acc, cmul(cload(cg + cgi * 2), cload(f + (q * G_ + g) * 2)));
    }
    cstore(cgf + idx * 2, acc);
}

// ---------------------------------------------------------------------------
// k1: filters[x,i,j,q,r] = embed(dX) + sum_{s,t<3} cgf[q,r,s,t]*dX[s]*dX[t]
//     layout: filters[(x*64+i)*1792 + j*28 + q*7 + r] (complex)
// ---------------------------------------------------------------------------
__global__ void k_filters(const float* __restrict__ dX, const float* __restrict__ cgf,
                          float* __restrict__ filters) {
    int tid = blockIdx.x * blockDim.x + threadIdx.x;
    const int total = X_ * N_ * N_ * Q_ * D_;   // 229376
    if (tid >= total) return;
    int r = tid % D_;
    int q = (tid / D_) % Q_;
    int p = tid / (Q_ * D_);                    // (x,i,j) point index
    const float* dx = dX + (long)p * 3 * 2;
    c32 dv[3] = { cload(dx), cload(dx + 2), cload(dx + 4) };
    c32 acc = {0.f, 0.f};
    if (q == 1 && r < 3) acc = dv[r];
    for (int s = 0; s < 3; ++s)
        for (int t = 0; t < 3; ++t) {
            c32 co = cload(cgf + (((q * D_ + r) * 3 + s) * 3 + t) * 2);
            acc = cadd(acc, cmul(co, cmul(dv[s], dv[t])));
        }
    cstore(filters + ((long)p * Q_ * D_ + q * D_ + r) * 2, acc);
}

// ---------------------------------------------------------------------------
// k2: U[xj,q,g,d,m,t] = sum_c V[xj,m,c,t] * W[q,c,g,d]
// ---------------------------------------------------------------------------
__global__ void k_U(const float* __restrict__ V, const float* __restrict__ W,
                    float* __restrict__ U) {
    long tid = (long)blockIdx.x * blockDim.x + threadIdx.x;
    const long total = (long)X_ * N_ * Q_ * G_ * C_ * Q_ * D_;   // 917504
    if (tid >= total) return;
    int t  = tid % D_;
    int m  = (tid / D_) % Q_;
    int d  = (tid / (D_ * Q_)) % C_;
    int g  = (tid / (D_ * Q_ * C_)) % G_;
    int q  = (tid / (D_ * Q_ * C_ * G_)) % Q_;
    long xj = tid / ((long)D_ * Q_ * C_ * G_ * Q_);              // x*64 + j
    c32 acc = {0.f, 0.f};
    for (int c = 0; c < C_; ++c) {
        c32 v = cload(V + (((xj * Q_ + m) * C_ + c) * D_ + t) * 2);
        c32 w = cload(W + ((((long)q * C_ + c) * G_ + g) * C_ + d) * 2);
        acc = cadd(acc, cmul(v, w));
    }
    cstore(U + tid * 2, acc);
}

// ---------------------------------------------------------------------------
// k3: Kmat[x, kidx(j,l,s), col(q,d,r)] = sum_{g,m,t} cg[g,l,s,m,t,q,r]*U[xj,q,g,d,m,t]
// ---------------------------------------------------------------------------
__global__ void k_K(const float* __restrict__ cg, const float* __restrict__ U,
                    float* __restrict__ Kmat) {
    long tid = (long)blockIdx.x * blockDim.x + threadIdx.x;
    const long total = (long)X_ * KDIM * NCOL;                   // 1605632
    if (tid >= total) return;
    int col  = tid % NCOL;
    int kidx = (tid / NCOL) % KDIM;
    int x    = (int)(tid / ((long)NCOL * KDIM));
    int q = col / (C_ * D_), d = (col / D_) % C_, r = col % D_;
    int j = kidx / (Q_ * D_), l = (kidx / D_) % Q_, s = kidx % D_;
    long xj = x * N_ + j;
    c32 acc = {0.f, 0.f};
    for (int g = 0; g < G_; ++g)
        for (int m = 0; m < Q_; ++m)
            for (int t = 0; t < D_; ++t) {
                long cgi = ((((((long)g * Q_ + l) * D_ + s) * Q_ + m) * D_ + t) * Q_ + q) * D_ + r;
                long ui  = ((((xj * Q_ + q) * G_ + g) * C_ + d) * Q_ + m) * D_ + t;
                acc = cadd(acc, cmul(cload(cg + cgi * 2), cload(U + ui * 2)));
            }
    cstore(Kmat + tid * 2, acc);
}

// ---------------------------------------------------------------------------
// k4: complex GEMM with FP32 WMMA, per x: Out[i,col] = sum_k F[i,k]*K[k,col]
//     Re = Fr*Kr + (-Fi)*Ki ; Im = Fr*Ki + Fi*Kr
//     One wave per 16x16 tile, v_wmma_f32_16x16x4_f32, EXEC all ones.
//     A frag (16x4): lane<16 -> K = kk+{0,1}; lane>=16 -> K = kk+{2,3}
//     B frag (4x16): same K split; column = lane&15
// ---------------------------------------------------------------------------
__global__ void k_gemm_wmma(const float* __restrict__ filters,
                            const float* __restrict__ Kmat,
                            float* __restrict__ out) {
    const int lane = threadIdx.x;
    const int mt = blockIdx.x;     // 0..3   (i tiles)
    const int nt = blockIdx.y;     // 0..27  (col tiles)
    const int x  = blockIdx.z;     // 0..1
    const int l15  = lane & 15;
    const int half = lane >> 4;
    const int mrow = mt * 16 + l15;
    const int ncol = nt * 16 + l15;

    v8f accRe = {0.f, 0.f, 0.f, 0.f, 0.f, 0.f, 0.f, 0.f};
    v8f accIm = {0.f, 0.f, 0.f, 0.f, 0.f, 0.f, 0.f, 0.f};

    const float* Abase = filters + (long)(x * N_ + mrow) * KDIM * 2;
    const float* Bbase = Kmat + (long)x * KDIM * NCOL * 2;

    for (int kk = 0; kk < KDIM; kk += 4) {
        const int k = kk + half * 2;
        // A: two consecutive complex values along K -> 16B aligned float4
        const float4 fa = *reinterpret_cast<const float4*>(Abase + (long)k * 2);
        v2f ar; ar[0] = fa.x;  ar[1] = fa.z;    // Re(F[k]), Re(F[k+1])
        v2f ai; ai[0] = fa.y;  ai[1] = fa.w;    // Im
        v2f an; an[0] = -fa.y; an[1] = -fa.w;   // -Im
        // B: rows k, k+1 at fixed column
        const float2 b0 = *reinterpret_cast<const float2*>(Bbase + ((long)k * NCOL + ncol) * 2);
        const float2 b1 = *reinterpret_cast<const float2*>(Bbase + ((long)(k + 1) * NCOL + ncol) * 2);
        v2f br; br[0] = b0.x; br[1] = b1.x;
        v2f bi; bi[0] = b0.y; bi[1] = b1.y;

        accRe = __builtin_amdgcn_wmma_f32_16x16x4_f32(false, ar, false, br, (short)0, accRe, false, false);
        accRe = __builtin_amdgcn_wmma_f32_16x16x4_f32(false, an, false, bi, (short)0, accRe, false, false);
        accIm = __builtin_amdgcn_wmma_f32_16x16x4_f32(false, ar, false, bi, (short)0, accIm, false, false);
        accIm = __builtin_amdgcn_wmma_f32_16x16x4_f32(false, ai, false, br, (short)0, accIm, false, false);
    }

    // D layout: VGPR rr -> M = rr + half*8, N = lane&15
    for (int rr = 0; rr < 8; ++rr) {
        const int i = mt * 16 + rr + half * 8;
        const long o = (((long)(x * N_ + i)) * NCOL + ncol) * 2;
        out[o]     = accRe[rr];
        out[o + 1] = accIm[rr];
    }
}

// ---------------------------------------------------------------------------
extern "C" void kernel_launch(void* const* d_in, const int* in_sizes, int n_in,
                              void* d_out, int out_size, void* d_ws, size_t ws_size,
                              hipStream_t stream) {
    (void)in_sizes; (void)n_in; (void)out_size; (void)ws_size;
    const float* dX = (const float*)d_in[0];
    const float* V  = (const float*)d_in[1];
    const float* W  = (const float*)d_in[2];
    const float* f  = (const float*)d_in[3];
    const float* cg = (const float*)d_in[4];
    // d_in[5] is l (==1); structure hardcoded accordingly.

    float* filters = (float*)d_ws;                 // 458752 floats ( 1.8 MB)
    float* U       = filters + 458752;             // 1835008 floats ( 7.3 MB)
    float* Kmat    = U + 1835008;                  // 3211264 floats (12.8 MB)
    float* cgf     = Kmat + 3211264;               // 504 floats
    float* out     = (float*)d_out;

    k_cgf<<<dim3(1), dim3(256), 0, stream>>>(f, cg, cgf);
    k_filters<<<dim3((229376 + 255) / 256), dim3(256), 0, stream>>>(dX, cgf, filters);
    k_U<<<dim3((917504 + 255) / 256), dim3(256), 0, stream>>>(V, W, U);
    k_K<<<dim3((1605632 + 255) / 256), dim3(256), 0, stream>>>(cg, U, Kmat);
    k_gemm_wmma<<<dim3(4, 28, 2), dim3(32), 0, stream>>>(filters, Kmat, out);
}